// PosDecoder_48533130445097
// MI455X (gfx1250) — compile-verified
//
#include <hip/hip_runtime.h>

// Problem constants (from the reference)
#define N_LOC   100001   // lookup table rows (row 0 is all zeros -> summing all rows == table[1:].sum)
#define FEAT    128      // features used
#define SRC_M   256      // src last-dim stride (only first 128 used)
#define OUT_N   1600     // 16 * 100 output scalars

typedef __attribute__((ext_vector_type(2))) float v2f;
typedef __attribute__((ext_vector_type(8))) float v8f;

// ---------------------------------------------------------------------------
// Kernel 1: streaming partial column-sums of the 100001 x 128 table.
// Memory-bound phase: 51.2 MB once -> ~2.2 us at 23.3 TB/s.
// Block = 256 threads = 8 waves; each iteration a block consumes 8 rows with
// fully-coalesced float4 (b128) loads. Two-stage deterministic reduction
// (LDS tree here, fixed-order cross-block sum in kernel 2) -> bit-identical
// results on every replay (no float atomics).
// ---------------------------------------------------------------------------
__global__ __launch_bounds__(256) void colsum_partial_kernel(
    const float* __restrict__ table, float* __restrict__ partial, int rows_per_block)
{
    __shared__ float4 red[256];
    const int t      = threadIdx.x;
    const int col4   = t & 31;   // float4 column 0..31  (covers 128 floats/row)
    const int rowoff = t >> 5;   // 0..7 row phase within the block
    long start = (long)blockIdx.x * rows_per_block;
    long end   = start + rows_per_block;
    if (end > N_LOC) end = N_LOC;

    float4 acc = make_float4(0.f, 0.f, 0.f, 0.f);
    for (long r = start + rowoff; r < end; r += 8) {
        const float4 v = ((const float4*)(table + (size_t)r * FEAT))[col4];
        acc.x += v.x; acc.y += v.y; acc.z += v.z; acc.w += v.w;
    }
    red[t] = acc;
    __syncthreads();

    if (rowoff == 0) {
        float4 s = red[col4];
        #pragma unroll
        for (int i = 1; i < 8; ++i) {
            const float4 v = red[i * 32 + col4];
            s.x += v.x; s.y += v.y; s.z += v.z; s.w += v.w;
        }
        ((float4*)partial)[(size_t)blockIdx.x * 32 + col4] = s;
    }
}

// ---------------------------------------------------------------------------
// Kernel 2: fixed-order reduction of per-block partials -> S[128], times sqrt(128).
// ---------------------------------------------------------------------------
__global__ __launch_bounds__(FEAT) void colsum_finalize_kernel(
    const float* __restrict__ partial, float* __restrict__ S, int nblk)
{
    const int t = threadIdx.x;   // 0..127
    float s = 0.f;
    for (int b = 0; b < nblk; ++b) s += partial[(size_t)b * FEAT + t];
    S[t] = s * 11.313708498984761f;   // sqrt(128)
}

// ---------------------------------------------------------------------------
// Kernel 3: out[i] = dot(src[i, 0:128], S) for i in 0..1599, via
// V_WMMA_F32_16X16X4_F32 (full fp32 precision).
// One wave32 per 16-row tile; 32 chained K=4 WMMAs over the 128-long K dim.
// B[k][n] = S[k] broadcast across all 16 N columns, so every column of D
// holds the matvec result; we write column N=0.
//
// ISA 7.12.2 layouts (wave32):
//   A (16x4 f32, 2 VGPRs): lanes 0-15 -> M=lane,   {K=0 (v0), K=1 (v1)}
//                          lanes 16-31 -> M=lane-16,{K=2 (v0), K=3 (v1)}
//   C/D (16x16 f32, 8 VGPRs): VGPR v, lanes 0-15 -> M=v, lanes 16-31 -> M=v+8,
//                             N = lane % 16.
// ---------------------------------------------------------------------------
__global__ __launch_bounds__(32) void matvec_wmma_kernel(
    const float* __restrict__ src, const float* __restrict__ S, float* __restrict__ out)
{
    const int lane = threadIdx.x;       // 0..31, EXEC all-ones for WMMA
    const int tile = blockIdx.x;        // 100 tiles of 16 rows
    const int half = lane >> 4;         // 0: K even-pair base, 1: K odd-pair base
    const int m    = lane & 15;

    const float* arow = src + (size_t)(tile * 16 + m) * SRC_M;

    v8f c = {};
    #pragma unroll
    for (int kc = 0; kc < FEAT / 4; ++kc) {
        const int kb = kc * 4 + half * 2;   // this lane holds K = kb, kb+1
        v2f a, b;
        a.x = arow[kb];  a.y = arow[kb + 1];
        b.x = S[kb];     b.y = S[kb + 1];   // B row K broadcast across N
        c = __builtin_amdgcn_wmma_f32_16x16x4_f32(
                /*neg_a=*/false, a, /*neg_b=*/false, b,
                /*c_mod=*/(short)0, c, /*reuse_a=*/false, /*reuse_b=*/false);
    }

    // Column N==0 lives in lanes 0 (rows M=0..7) and 16 (rows M=8..15).
    if (m == 0) {
        float* o = out + tile * 16 + half * 8;
        #pragma unroll
        for (int v = 0; v < 8; ++v) o[v] = c[v];
    }
}

// ---------------------------------------------------------------------------
// Launcher. Inputs (reference setup_inputs order):
//   d_in[0] = src          float32 [16,100,256]
//   d_in[1] = ds           int64   [16,100]   (unused by the reference math)
//   d_in[2] = lookup_table float32 [100001,128]
// d_out: 1600 float32.
// ---------------------------------------------------------------------------
extern "C" void kernel_launch(void* const* d_in, const int* in_sizes, int n_in,
                              void* d_out, int out_size, void* d_ws, size_t ws_size,
                              hipStream_t stream) {
    (void)in_sizes; (void)n_in; (void)out_size;
    const float* src   = (const float*)d_in[0];
    const float* table = (const float*)d_in[2];
    float*       out   = (float*)d_out;

    // Workspace layout: [nblk * 128 partial sums][128-float S vector]
    int nblk = 512;
    while (nblk > 1 &&
           ((size_t)nblk * FEAT + FEAT) * sizeof(float) > ws_size)
        nblk >>= 1;
    float* partial = (float*)d_ws;
    float* Svec    = partial + (size_t)nblk * FEAT;
    const int rows_per_block = (N_LOC + nblk - 1) / nblk;

    colsum_partial_kernel <<<nblk,       256,  0, stream>>>(table, partial, rows_per_block);
    colsum_finalize_kernel<<<1,          FEAT, 0, stream>>>(partial, Svec, nblk);
    matvec_wmma_kernel    <<<OUT_N / 16, 32,   0, stream>>>(src, Svec, out);
}